// GroupSoftThresholdingLayer_15023795601934
// MI455X (gfx1250) — compile-verified
//
#include <hip/hip_runtime.h>

typedef __attribute__((ext_vector_type(2))) float v2f;
typedef __attribute__((ext_vector_type(4))) float v4f;
typedef __attribute__((ext_vector_type(8))) float v8f;

#define NB1 1024        // reduction blocks (contiguous 64-row chunks)
#define NB3 4096        // scaling blocks (contiguous chunks, reversed order)
#define EPS 1e-7f

#if defined(__HIP_DEVICE_COMPILE__) && __has_builtin(__builtin_amdgcn_wmma_f32_16x16x4_f32)
#define HAVE_WMMA_4XF32 1
#else
#define HAVE_WMMA_4XF32 0
#endif

// lds[256] holds per-thread partials; group g's 4 partials are lds[4g..4g+3].
// WMMA path: A = all-ones 16x4, B packs 16 groups' quads per wave (waves 0..3),
// D[m][n] = sum_k B[k][n] = group sum (replicated over m) -> d[0], lanes 0..15.
// With A==1 the K-slot permutation inside B's layout is irrelevant; only the
// (lane&15)->column mapping matters, which holds for the documented layouts.
__device__ __forceinline__ bool group_sums(const float* lds, int t, int* gOut, float* sOut) {
#if HAVE_WMMA_4XF32
    const int wave = t >> 5;
    const int lane = t & 31;
    if (wave < 4) {                       // wave-uniform branch: EXEC all ones inside
        v2f a; a[0] = 1.0f; a[1] = 1.0f;  // ones matrix, layout-independent
        const int base = (wave << 6) + ((lane & 15) << 2) + ((lane >> 4) << 1);
        v2f b; b[0] = lds[base]; b[1] = lds[base + 1];
        v8f c = {};
        v8f d = __builtin_amdgcn_wmma_f32_16x16x4_f32(
            false, a, false, b, (short)0, c, false, false);
        if (lane < 16) { *gOut = (wave << 4) + lane; *sOut = d[0]; return true; }
    }
    return false;
#else
    if (t < 64) {
        *gOut = t;
        *sOut = (lds[4 * t] + lds[4 * t + 1]) + (lds[4 * t + 2] + lds[4 * t + 3]);
        return true;
    }
    return false;
#endif
}

// Stage 1: per-block partial sum of squares per group over a CONTIGUOUS chunk of
// rows (ascending block order => high addresses are the L2-resident tail after
// this kernel). Row = 1024 floats = 256 float4; thread t owns channels 4t..4t+3
// (group t>>2). 8-row unroll => 8 independent b128 loads in flight per thread.
// Default (RT) loads on purpose: we WANT x retained in L2 for stage 3.
__global__ void __launch_bounds__(256) k_partial(const float* __restrict__ x,
                                                 float* __restrict__ part,
                                                 int rows, int nb) {
    const int t = threadIdx.x;
    const v4f* __restrict__ x4 = (const v4f*)x;
    const int rpb = (rows + nb - 1) / nb;           // rows per block (64)
    const int rBeg = blockIdx.x * rpb;
    const int rEnd = (rBeg + rpb < rows) ? (rBeg + rpb) : rows;

    float acc8[8] = {0.f, 0.f, 0.f, 0.f, 0.f, 0.f, 0.f, 0.f};
    for (int r0 = rBeg; r0 < rEnd; r0 += 8) {
        v4f v[8];
#pragma unroll
        for (int u = 0; u < 8; ++u)
            if (r0 + u < rEnd) v[u] = x4[(size_t)(r0 + u) * 256 + t];
#pragma unroll
        for (int u = 0; u < 8; ++u)
            if (r0 + u < rEnd) {
                acc8[u] += fmaf(v[u].x, v[u].x, fmaf(v[u].y, v[u].y,
                           fmaf(v[u].z, v[u].z, v[u].w * v[u].w)));
            }
    }
    float acc = ((acc8[0] + acc8[1]) + (acc8[2] + acc8[3])) +
                ((acc8[4] + acc8[5]) + (acc8[6] + acc8[7]));

    __shared__ float lds[256];
    lds[t] = acc;
    __syncthreads();
    int g; float s;
    if (group_sums(lds, t, &g, &s))
        part[(size_t)blockIdx.x * 64 + g] = s;
}

// Stage 2: fold NB1 block-partials per group (fixed order => deterministic),
// WMMA for the final x4 fold, then scale[g] = relu(sqrt(s+eps)-theta)/sqrt(s+eps).
__global__ void __launch_bounds__(256) k_finalize(const float* __restrict__ part,
                                                  const float* __restrict__ theta,
                                                  float* __restrict__ scale, int nb) {
    const int t = threadIdx.x;
    const int g = t >> 2, slot = t & 3;
    const int chunk = nb >> 2;                      // nb divisible by 4
    float acc = 0.f;
    for (int i = slot * chunk; i < (slot + 1) * chunk; ++i)
        acc += part[(size_t)i * 64 + g];

    __shared__ float lds[256];
    lds[t] = acc;
    __syncthreads();
    int gg; float s;
    if (group_sums(lds, t, &gg, &s)) {
        float nrm = sqrtf(s + EPS);
        float r = nrm - theta[gg];
        r = r > 0.f ? r : 0.f;
        scale[gg] = r / nrm;
    }
}

// Stage 3: out = x * scale[group]. Contiguous chunks traversed in REVERSE global
// order so the first reads hit the L2-resident tail of x left by stage 1.
// x loads: nontemporal (last use, don't retain). out stores: nontemporal
// (write-once, don't evict x from L2). Every slab base is a multiple of 256
// float4, so each thread's group (t>>2) is constant -> scale loaded once.
__global__ void __launch_bounds__(256) k_scale(const float* __restrict__ x,
                                               const float* __restrict__ scale,
                                               float* __restrict__ out, long long n4) {
    const int t = threadIdx.x;
    const v4f* __restrict__ x4 = (const v4f*)x;
    v4f* __restrict__ o4 = (v4f*)out;
    const size_t N = (size_t)n4;                    // N % 256 == 0 (D = 1024)
    const size_t nblk = gridDim.x;
    const size_t c = nblk - 1 - blockIdx.x;         // reversed chunk order
    const size_t slabsTotal = N >> 8;               // 256-float4 slabs
    const size_t spb = (slabsTotal + nblk - 1) / nblk;
    const long sBeg = (long)(c * spb);
    long sEnd = (long)(c * spb + spb);
    if ((size_t)sEnd > slabsTotal) sEnd = (long)slabsTotal;
    if (sBeg >= sEnd) return;

    const float sc = scale[t >> 2];

    long s = sEnd;
    // Peel so the remaining span is a multiple of 4 slabs (topmost first).
    while (((s - sBeg) & 3) != 0) {
        --s;
        const size_t v = ((size_t)s << 8) + t;
        v4f d = __builtin_nontemporal_load(x4 + v);
        d.x *= sc; d.y *= sc; d.z *= sc; d.w *= sc;
        __builtin_nontemporal_store(d, o4 + v);
    }
    // Bulk: quads of slabs, descending (4 independent b128 loads per thread).
    for (s -= 4; s >= sBeg; s -= 4) {
        const size_t v = ((size_t)s << 8) + t;
        v4f d0 = __builtin_nontemporal_load(x4 + v);
        v4f d1 = __builtin_nontemporal_load(x4 + v + 256);
        v4f d2 = __builtin_nontemporal_load(x4 + v + 512);
        v4f d3 = __builtin_nontemporal_load(x4 + v + 768);
        d0.x *= sc; d0.y *= sc; d0.z *= sc; d0.w *= sc;
        d1.x *= sc; d1.y *= sc; d1.z *= sc; d1.w *= sc;
        d2.x *= sc; d2.y *= sc; d2.z *= sc; d2.w *= sc;
        d3.x *= sc; d3.y *= sc; d3.z *= sc; d3.w *= sc;
        __builtin_nontemporal_store(d0, o4 + v);
        __builtin_nontemporal_store(d1, o4 + v + 256);
        __builtin_nontemporal_store(d2, o4 + v + 512);
        __builtin_nontemporal_store(d3, o4 + v + 768);
    }
}

extern "C" void kernel_launch(void* const* d_in, const int* in_sizes, int n_in,
                              void* d_out, int out_size, void* d_ws, size_t ws_size,
                              hipStream_t stream) {
    const float* x     = (const float*)d_in[0];   // (32, 2048, 1024) fp32
    const float* theta = (const float*)d_in[1];   // (64,) fp32
    float* out = (float*)d_out;

    const long long n = (long long)in_sizes[0];   // 67,108,864
    const int rows = (int)(n / 1024);             // 65,536 rows of D=1024

    float* part  = (float*)d_ws;                  // [NB1][64] partials (fully overwritten)
    float* scale = part + (size_t)NB1 * 64;       // [64] group scales (fully overwritten)

    k_partial <<<NB1, 256, 0, stream>>>(x, part, rows, NB1);
    k_finalize<<<1,   256, 0, stream>>>(part, theta, scale, NB1);
    k_scale   <<<NB3, 256, 0, stream>>>(x, scale, out, n / 4);
}